// CustomDFAModel_16423954939991
// MI455X (gfx1250) — compile-verified
//
#include <hip/hip_runtime.h>

// Problem constants (fixed by the reference)
#define BS_      2
#define ANCH_    900
#define PTS_     13
#define CAMS_    6
#define LVLS_    4
#define EMBED_   256
#define GROUPS_  8
#define PER_CAM_ 14960                    // 64*176 + 32*88 + 16*44 + 8*22
#define NPL_     (PTS_ * CAMS_)           // 78 samples per level
#define NSAMP_   (LVLS_ * NPL_)           // 312 samples per anchor

__global__ __launch_bounds__(256)
void dfa_gather_kernel(const float* __restrict__ value,
                       const int*   __restrict__ shapes,   // (CAMS, LVLS, 2), cam-tiled
                       const int*   __restrict__ starts,   // (CAMS, LVLS),   cam-tiled
                       const float* __restrict__ loc,      // (BS, A, P, CAMS, 2)
                       const float* __restrict__ aw,       // (BS, A, P, CAMS, LVLS, GROUPS)
                       float*       __restrict__ out)      // (BS, A, EMBED)
{
    __shared__ __align__(16) int    sIdx[NSAMP_ * 4];       // corner row indices into value
    __shared__ __align__(16) float  sBw [NSAMP_ * 4];       // bilinear weight * validity
    __shared__ __align__(16) float  sW  [NSAMP_ * GROUPS_]; // attention weight per group
    __shared__ __align__(16) float4 sPart[4][64];           // cross-phase reduction

    const int tid = threadIdx.x;
    const int a   = blockIdx.x % ANCH_;
    const int b   = blockIdx.x / ANCH_;

    // ---- Phase 1: per-sample metadata precompute into LDS -------------------
    for (int s = tid; s < NSAMP_; s += 256) {
        const int lvl = s / NPL_;
        const int r   = s - lvl * NPL_;
        const int p   = r / CAMS_;
        const int c   = r - p * CAMS_;
        const int H   = shapes[lvl * 2 + 0];
        const int W   = shapes[lvl * 2 + 1];
        const int st  = starts[lvl];

        const float* lp = loc + ((((size_t)b * ANCH_ + a) * PTS_ + p) * CAMS_ + c) * 2;
        const float lw = lp[0];
        const float lh = lp[1];
        const float hf = lh * (float)H - 0.5f;
        const float wf = lw * (float)W - 0.5f;
        const float h0 = floorf(hf);
        const float w0 = floorf(wf);
        const float dh = hf - h0;
        const float dw = wf - w0;
        const int   h0i  = (int)h0;
        const int   w0i  = (int)w0;
        const int   base = (b * CAMS_ + c) * PER_CAM_ + st;

        #pragma unroll
        for (int k = 0; k < 4; ++k) {
            const int  hi = h0i + (k >> 1);
            const int  wi = w0i + (k & 1);
            const bool valid = (hi >= 0) & (hi < H) & (wi >= 0) & (wi < W);
            const float bwh = (k >> 1) ? dh : (1.0f - dh);
            const float bww = (k & 1)  ? dw : (1.0f - dw);
            const int  hc = min(max(hi, 0), H - 1);
            const int  wc = min(max(wi, 0), W - 1);
            sIdx[s * 4 + k] = base + hc * W + wc;
            sBw [s * 4 + k] = valid ? (bwh * bww) : 0.0f;
        }

        const float4* wp = (const float4*)(aw +
            (((((size_t)b * ANCH_ + a) * PTS_ + p) * CAMS_ + c) * LVLS_ + lvl) * GROUPS_);
        float4* swv = (float4*)(&sW[s * GROUPS_]);
        swv[0] = wp[0];
        swv[1] = wp[1];
    }
    __syncthreads();

    // ---- Phase 2: coalesced b128 gathers + weighted accumulate --------------
    // Thread layout: 64 channel-quads x 4 sample-phases.
    const int q   = tid & 63;   // channel quad: channels [4q, 4q+4)
    const int sc  = tid >> 6;   // sample phase 0..3
    const int ch0 = q << 2;
    const int g   = q >> 3;     // group of this quad (32 channels/group)

    const float* vbase = value + ch0;   // fold channel offset into base pointer

    float4 accA = make_float4(0.f, 0.f, 0.f, 0.f);
    float4 accB = make_float4(0.f, 0.f, 0.f, 0.f);
    const int4*   idxv = (const int4*)sIdx;
    const float4* bwv  = (const float4*)sBw;

    // 78 samples per thread, 2x unrolled -> 39 iterations, 8 b128 loads in flight.
    for (int s = sc; s < NSAMP_; s += 8) {
        const int sB = s + 4;

        const int4   idA = idxv[s];
        const int4   idB = idxv[sB];
        const float4 bwA = bwv[s];
        const float4 bwB = bwv[sB];
        const float  wgA = sW[s  * GROUPS_ + g];
        const float  wgB = sW[sB * GROUPS_ + g];

        // Prefetch the next unrolled iteration's 8 rows (distance = 8 samples).
        // Guard s+8 < NSAMP_ also bounds s+12 (max 311 < 312).
        if (s + 8 < NSAMP_) {
            const int4 ip0 = idxv[s + 8];
            const int4 ip1 = idxv[s + 12];
            __builtin_prefetch(vbase + (size_t)ip0.x * EMBED_, 0, 1);
            __builtin_prefetch(vbase + (size_t)ip0.y * EMBED_, 0, 1);
            __builtin_prefetch(vbase + (size_t)ip0.z * EMBED_, 0, 1);
            __builtin_prefetch(vbase + (size_t)ip0.w * EMBED_, 0, 1);
            __builtin_prefetch(vbase + (size_t)ip1.x * EMBED_, 0, 1);
            __builtin_prefetch(vbase + (size_t)ip1.y * EMBED_, 0, 1);
            __builtin_prefetch(vbase + (size_t)ip1.z * EMBED_, 0, 1);
            __builtin_prefetch(vbase + (size_t)ip1.w * EMBED_, 0, 1);
        }

        // 8 independent 128-bit gathers
        const float4 a0 = *(const float4*)(vbase + (size_t)idA.x * EMBED_);
        const float4 a1 = *(const float4*)(vbase + (size_t)idA.y * EMBED_);
        const float4 a2 = *(const float4*)(vbase + (size_t)idA.z * EMBED_);
        const float4 a3 = *(const float4*)(vbase + (size_t)idA.w * EMBED_);
        const float4 b0v = *(const float4*)(vbase + (size_t)idB.x * EMBED_);
        const float4 b1v = *(const float4*)(vbase + (size_t)idB.y * EMBED_);
        const float4 b2v = *(const float4*)(vbase + (size_t)idB.z * EMBED_);
        const float4 b3v = *(const float4*)(vbase + (size_t)idB.w * EMBED_);

        const float cA0 = bwA.x * wgA;
        const float cA1 = bwA.y * wgA;
        const float cA2 = bwA.z * wgA;
        const float cA3 = bwA.w * wgA;
        const float cB0 = bwB.x * wgB;
        const float cB1 = bwB.y * wgB;
        const float cB2 = bwB.z * wgB;
        const float cB3 = bwB.w * wgB;

        accA.x += cA0 * a0.x + cA1 * a1.x + cA2 * a2.x + cA3 * a3.x;
        accA.y += cA0 * a0.y + cA1 * a1.y + cA2 * a2.y + cA3 * a3.y;
        accA.z += cA0 * a0.z + cA1 * a1.z + cA2 * a2.z + cA3 * a3.z;
        accA.w += cA0 * a0.w + cA1 * a1.w + cA2 * a2.w + cA3 * a3.w;

        accB.x += cB0 * b0v.x + cB1 * b1v.x + cB2 * b2v.x + cB3 * b3v.x;
        accB.y += cB0 * b0v.y + cB1 * b1v.y + cB2 * b2v.y + cB3 * b3v.y;
        accB.z += cB0 * b0v.z + cB1 * b1v.z + cB2 * b2v.z + cB3 * b3v.z;
        accB.w += cB0 * b0v.w + cB1 * b1v.w + cB2 * b2v.w + cB3 * b3v.w;
    }

    accA.x += accB.x;
    accA.y += accB.y;
    accA.z += accB.z;
    accA.w += accB.w;

    sPart[sc][q] = accA;
    __syncthreads();

    // ---- Phase 3: combine the 4 sample phases, write output -----------------
    if (tid < 64) {
        float4       r0 = sPart[0][tid];
        const float4 r1 = sPart[1][tid];
        const float4 r2 = sPart[2][tid];
        const float4 r3 = sPart[3][tid];
        r0.x += r1.x + r2.x + r3.x;
        r0.y += r1.y + r2.y + r3.y;
        r0.z += r1.z + r2.z + r3.z;
        r0.w += r1.w + r2.w + r3.w;
        *(float4*)(out + ((size_t)b * ANCH_ + a) * EMBED_ + (tid << 2)) = r0;
    }
}

extern "C" void kernel_launch(void* const* d_in, const int* in_sizes, int n_in,
                              void* d_out, int out_size, void* d_ws, size_t ws_size,
                              hipStream_t stream)
{
    const float* value  = (const float*)d_in[0];
    const int*   shapes = (const int*)  d_in[1];
    const int*   starts = (const int*)  d_in[2];
    const float* loc    = (const float*)d_in[3];
    const float* aw     = (const float*)d_in[4];
    float*       out    = (float*)d_out;

    dim3 grid(BS_ * ANCH_);   // one workgroup per (batch, anchor)
    dim3 block(256);          // 8 wave32s
    hipLaunchKernelGGL(dfa_gather_kernel, grid, block, 0, stream,
                       value, shapes, starts, loc, aw, out);
}